// KoopmanLQR_8529805050432
// MI455X (gfx1250) — compile-verified
//
#include <hip/hip_runtime.h>

// Batched finite-horizon LQR backward Riccati recursion for MI455X (gfx1250).
// fp32 throughout (50-step recursion is numerically sensitive) using the native
// CDNA5 fp32 matrix op V_WMMA_F32_16X16X4_F32 with RTxCT register-blocked
// macro-tiles per wave to amortize fragment loads across WMMAs.
//
// One workgroup (256 threads = 8 wave32) per batch element. Carry V and A-BK
// live in global scratch (L2-resident, 32MB total); the 256KB intermediate
// W = V@ABK lives fully in LDS (CDNA5: 320KB/WGP).

#define BATCHN 64
#define KDIM   256
#define UDIM   64
#define TH     50

typedef float v2f __attribute__((ext_vector_type(2)));
typedef float v4f __attribute__((ext_vector_type(4)));
typedef float v8f __attribute__((ext_vector_type(8)));

__device__ __forceinline__ v2f ld2(const float* p) { return *(const v2f*)p; }

// Row-major WMMA GEMM:  D = (NEG ? C - A*B : C + A*B), C present iff HAS_C.
// TA: A operand stored transposed (element (r,c) read at Pa[c*lda + r]).
// M,N multiples of 16*RT / 16*CT resp.; Kd multiple of 4.
// Each wave owns RTxCT 16x16 output tiles: per K-step RT A-frags + CT B-frags
// feed RT*CT WMMAs. Macro-tiles striped across the 8 waves; loops are wave-
// uniform so EXEC is all-ones at every WMMA (ISA requirement).
template <int RT, int CT, bool TA, bool HAS_C, bool NEG>
__device__ __forceinline__ void tile_gemm(const float* __restrict__ Pa, int lda,
                                          const float* __restrict__ Pb, int ldb,
                                          const float* __restrict__ Pc, int ldc,
                                          float* __restrict__ Pd, int ldd,
                                          int M, int N, int Kd)
{
    const int lane = threadIdx.x & 31;
    const int wave = threadIdx.x >> 5;
    const int gM = (M >> 4) / RT;
    const int gN = (N >> 4) / CT;
    // Lane layout per ISA 7.12.2 (wave32, f32 16x16x4):
    //   A frag: VGPR0/1 = A[m][k + 2*(lane/16) + {0,1}],  m = lane%16
    //   B frag: VGPR0/1 = B[k + 2*(lane/16) + {0,1}][n],  n = lane%16
    //   C/D:    VGPR r  = D[8*(lane/16) + r][n]
    const int m  = lane & 15;              // also n for B/C/D operands
    const int kk = (lane >> 4) << 1;
    const int mb = (lane >> 4) << 3;

    for (int g = wave; g < gM * gN; g += 8) {
        const int row0 = (g / gN) * (RT * 16);
        const int col0 = (g % gN) * (CT * 16);

        v8f acc[RT][CT];
#pragma unroll
        for (int i = 0; i < RT; ++i)
#pragma unroll
            for (int j = 0; j < CT; ++j)
                acc[i][j] = (v8f){0.f, 0.f, 0.f, 0.f, 0.f, 0.f, 0.f, 0.f};

        for (int k0 = 0; k0 < Kd; k0 += 4) {
            v2f a[RT], b[CT];
#pragma unroll
            for (int i = 0; i < RT; ++i) {
                const int r = row0 + i * 16 + m;
                if (!TA) {
                    a[i] = ld2(Pa + r * lda + (k0 + kk));      // one b64 load
                } else {
                    a[i].x = Pa[(k0 + kk) * lda + r];
                    a[i].y = Pa[(k0 + kk + 1) * lda + r];
                }
            }
#pragma unroll
            for (int j = 0; j < CT; ++j) {
                const int c = col0 + j * 16 + m;
                b[j].x = Pb[(k0 + kk) * ldb + c];
                b[j].y = Pb[(k0 + kk + 1) * ldb + c];
            }
#pragma unroll
            for (int i = 0; i < RT; ++i)
#pragma unroll
                for (int j = 0; j < CT; ++j)
                    // 8 args: (neg_a, A, neg_b, B, c_mod, C, reuse_a, reuse_b)
                    acc[i][j] = __builtin_amdgcn_wmma_f32_16x16x4_f32(
                        false, a[i], false, b[j], (short)0, acc[i][j], false, false);
        }

#pragma unroll
        for (int i = 0; i < RT; ++i)
#pragma unroll
            for (int j = 0; j < CT; ++j) {
#pragma unroll
                for (int r = 0; r < 8; ++r) {
                    const int rr = row0 + i * 16 + mb + r;
                    const int cc = col0 + j * 16 + m;
                    float v = acc[i][j][r];
                    if (HAS_C) {
                        const float cv = Pc[rr * ldc + cc];
                        v = NEG ? (cv - v) : (cv + v);
                    } else if (NEG) {
                        v = -v;
                    }
                    Pd[rr * ldd + cc] = v;
                }
            }
    }
}

__global__ __launch_bounds__(256)
void lqr_riccati_kernel(const float* __restrict__ A, const float* __restrict__ Bm,
                        const float* __restrict__ Q, const float* __restrict__ R,
                        const float* __restrict__ goals, float* __restrict__ out,
                        float* __restrict__ ws)
{
    // 256KB aliased LDS arena + small vectors (~258.7KB total, < 320KB/WGP).
    __shared__ float smem[KDIM * KDIM];
    __shared__ float sv[KDIM], svn[KDIM];
    __shared__ float sbv[UDIM], skg[UDIM], spcol[UDIM];

    float* sBtV = smem;                    // [U][K]   64KB  (later reused as Kg)
    float* sT1  = smem + UDIM * KDIM;      // [U][K]   64KB
    float* sG   = smem + 2 * UDIM * KDIM;  // [64][128] 32KB (Gauss-Jordan [Vuu|I])
    float* sW   = smem;                    // [K][K]  256KB  (phase 2, overlays all)

    const int b   = blockIdx.x;
    const int tid = threadIdx.x;

    const float* Ab = A     + (size_t)b * KDIM * KDIM;
    const float* Bb = Bm    + (size_t)b * KDIM * UDIM;
    const float* Qb = Q     + (size_t)b * KDIM * KDIM;
    const float* Rb = R     + (size_t)b * UDIM * UDIM;
    const float* Gb = goals + (size_t)b * (TH + 1) * KDIM;

    float* gV   = ws + (size_t)b * KDIM * KDIM;                                // carry V
    float* gABK = ws + (size_t)BATCHN * KDIM * KDIM + (size_t)b * KDIM * KDIM; // A - B*Kg

    // ---- init: V0 = Q ; v0 = Q @ goals[:, T] ----
    for (int i = tid; i < (KDIM * KDIM) / 4; i += 256)
        ((v4f*)gV)[i] = ((const v4f*)Qb)[i];
    {
        const float* gT = Gb + TH * KDIM;
        const float* qr = Qb + tid * KDIM;
        float s = 0.f;
        for (int j = 0; j < KDIM; ++j) s += qr[j] * gT[j];
        sv[tid] = s;
    }
    __syncthreads();

    for (int t = TH - 1; t >= 0; --t) {
        // 1. BtV = B^T @ V                              [U,K]
        tile_gemm<4, 2, true, false, false>(Bb, UDIM, gV, KDIM, nullptr, 0,
                                            sBtV, KDIM, UDIM, KDIM, KDIM);
        __syncthreads();

        // 2. Vuu = BtV @ B + R  -> sG[:, :64];  identity -> sG[:, 64:]
        tile_gemm<1, 2, false, true, false>(sBtV, KDIM, Bb, UDIM, Rb, UDIM,
                                            sG, 128, UDIM, UDIM, KDIM);
        for (int i = tid; i < UDIM * UDIM; i += 256) {
            const int r = i >> 6, c = i & 63;
            sG[r * 128 + 64 + c] = (r == c) ? 1.0f : 0.0f;
        }
        __syncthreads();

        // 3. Gauss-Jordan inverse of SPD Vuu (no pivoting needed).
        for (int p = 0; p < UDIM; ++p) {
            if (tid < UDIM) spcol[tid] = sG[tid * 128 + p];
            __syncthreads();
            if (tid < 128) sG[p * 128 + tid] *= (1.0f / spcol[p]);
            __syncthreads();
            {
                const int r = tid >> 2;
                if (r != p) {
                    const float f = spcol[r];
                    const float* prow = sG + p * 128;
                    float* rrow = sG + r * 128;
                    for (int c = tid & 3; c < 128; c += 4) rrow[c] -= f * prow[c];
                }
            }
            __syncthreads();
        }
        // Vuu_inv now lives at sG[:, 64:128] (ld = 128)

        // 4. T1 = Vuu_inv @ BtV                         [U,K], Kd=U
        tile_gemm<4, 2, false, false, false>(sG + 64, 128, sBtV, KDIM, nullptr, 0,
                                             sT1, KDIM, UDIM, KDIM, UDIM);
        __syncthreads();

        // 5. Kg = T1 @ A  (into sBtV)                   [U,K]
        tile_gemm<4, 2, false, false, false>(sT1, KDIM, Ab, KDIM, nullptr, 0,
                                             sBtV, KDIM, UDIM, KDIM, KDIM);
        // 6a. bv = B^T v  (overlaps step 5, disjoint buffers)
        if (tid < UDIM) {
            float s = 0.f;
            for (int j = 0; j < KDIM; ++j) s += Bb[j * UDIM + tid] * sv[j];
            sbv[tid] = s;
        }
        __syncthreads();
        // 6b. kg = Vuu_inv @ bv
        if (tid < UDIM) {
            const float* irow = sG + tid * 128 + 64;
            float s = 0.f;
            for (int u = 0; u < UDIM; ++u) s += irow[u] * sbv[u];
            skg[tid] = s;
        }
        __syncthreads();

        // 7. emit gains: out[t][b][u][:K]=Kg, out[t][b][u][K]=kg
        for (int idx = tid; idx < UDIM * (KDIM + 1); idx += 256) {
            const int u = idx / (KDIM + 1);
            const int k = idx % (KDIM + 1);
            const float val = (k < KDIM) ? sBtV[u * KDIM + k] : skg[u];
            out[(((size_t)t * BATCHN + b) * UDIM + u) * (KDIM + 1) + k] = val;
        }

        // 8. ABK = A - B @ Kg                           [K,K] -> global scratch
        tile_gemm<4, 2, false, true, true>(Bb, UDIM, sBtV, KDIM, Ab, KDIM,
                                           gABK, KDIM, KDIM, KDIM, UDIM);
        __syncthreads();

        // 9. W = V @ ABK  (entire 256x256 result in LDS)
        tile_gemm<4, 2, false, false, false>(gV, KDIM, gABK, KDIM, nullptr, 0,
                                             sW, KDIM, KDIM, KDIM, KDIM);
        __syncthreads();

        // 10. V' = A^T @ W + Q   (overwrites carry V)
        tile_gemm<4, 2, true, true, false>(Ab, KDIM, sW, KDIM, Qb, KDIM,
                                           gV, KDIM, KDIM, KDIM, KDIM);

        // 11. v' = ABK^T v + Q @ g_t   (one component per thread)
        {
            float s = 0.f;
            for (int j = 0; j < KDIM; ++j) s += gABK[j * KDIM + tid] * sv[j];
            const float* g  = Gb + t * KDIM;
            const float* qr = Qb + tid * KDIM;
            for (int j = 0; j < KDIM; ++j) s += qr[j] * g[j];
            svn[tid] = s;
        }
        __syncthreads();
        sv[tid] = svn[tid];
        __syncthreads();
    }
}

extern "C" void kernel_launch(void* const* d_in, const int* in_sizes, int n_in,
                              void* d_out, int out_size, void* d_ws, size_t ws_size,
                              hipStream_t stream)
{
    (void)in_sizes; (void)n_in; (void)out_size; (void)ws_size;
    const float* A     = (const float*)d_in[0];
    const float* Bm    = (const float*)d_in[1];
    const float* Q     = (const float*)d_in[2];
    const float* R     = (const float*)d_in[3];
    const float* goals = (const float*)d_in[4];

    lqr_riccati_kernel<<<BATCHN, 256, 0, stream>>>(
        A, Bm, Q, R, goals, (float*)d_out, (float*)d_ws);
}